// collisionAvoidanceModel_61134564491438
// MI455X (gfx1250) — compile-verified
//
#include <hip/hip_runtime.h>
#include <hip/hip_bf16.h>

typedef float v2f __attribute__((ext_vector_type(2)));
typedef float v8f __attribute__((ext_vector_type(8)));

// ---------------------------------------------------------------------------
// Ordered-uint encoding of float so unsigned atomicMax == float max.
// ---------------------------------------------------------------------------
__device__ __forceinline__ unsigned fenc(float f) {
    unsigned u = __float_as_uint(f);
    return (u & 0x80000000u) ? ~u : (u | 0x80000000u);
}
__device__ __forceinline__ float fdec(unsigned u) {
    return __uint_as_float((u & 0x80000000u) ? (u & 0x7fffffffu) : ~u);
}

// ---------------------------------------------------------------------------
// Elementwise: min-max normalize  y = (x - lb) / (ub - lb),  C_IN = 16
// ---------------------------------------------------------------------------
__global__ void k_norm(const float* __restrict__ x,
                       const float* __restrict__ lb,
                       const float* __restrict__ ub,
                       float* __restrict__ y, int total) {
    int i = blockIdx.x * blockDim.x + threadIdx.x;
    if (i >= total) return;
    int c = i & 15;
    y[i] = (x[i] - lb[c]) / (ub[c] - lb[c]);
}

// ---------------------------------------------------------------------------
// FP32 WMMA GEMM:  out[n,M] = A[n,K] @ W[K,M] + bias[M]
// One wave (32 lanes) computes one 16x16 tile via V_WMMA_F32_16X16X4_F32.
// blockDim = (32,4): 4 waves handle 4 adjacent column tiles.
// Requires n % 16 == 0, M % 64 == 0, K % 4 == 0 (true for all call sites).
// ---------------------------------------------------------------------------
__global__ __launch_bounds__(128) void k_gemm_wmma(
    const float* __restrict__ A, const float* __restrict__ W,
    const float* __restrict__ bias, float* __restrict__ out,
    int K, int M) {
    const int lane  = threadIdx.x;                       // 0..31
    const int tn    = blockIdx.x * 4 + threadIdx.y;      // column tile
    const int tm    = blockIdx.y;                        // row tile
    const int l15   = lane & 15;
    const int hi    = lane >> 4;                         // 0 or 1
    const int khalf = hi * 2;                            // K sub-offset 0 / 2

    const int row = tm * 16 + l15;
    const int col = tn * 16 + l15;

    const float* __restrict__ arow = A + (size_t)row * K;
    v8f acc = {0.f, 0.f, 0.f, 0.f, 0.f, 0.f, 0.f, 0.f};

    for (int k = 0; k < K; k += 4) {
        v2f a, b;
        // A 16x4: lanes 0-15 -> K = k+0/k+1 of row; lanes 16-31 -> K = k+2/k+3
        a.x = arow[k + khalf];
        a.y = arow[k + khalf + 1];
        // B 4x16: lanes 0-15 -> K = k+0/k+1 of col; lanes 16-31 -> K = k+2/k+3
        b.x = W[(size_t)(k + khalf) * M + col];
        b.y = W[(size_t)(k + khalf + 1) * M + col];
        acc = __builtin_amdgcn_wmma_f32_16x16x4_f32(
            /*neg_a=*/false, a, /*neg_b=*/false, b,
            /*c_mod=*/(short)0, acc, /*reuse_a=*/false, /*reuse_b=*/false);
    }

    // C/D layout: VGPR r holds M = r + 8*hi, N = lane&15
    const float bc = bias[col];
    float* __restrict__ o = out + (size_t)(tm * 16 + hi * 8) * M + col;
#pragma unroll
    for (int r = 0; r < 8; ++r)
        o[(size_t)r * M] = acc[r] + bc;
}

// ---------------------------------------------------------------------------
// Edge pass A: one WAVE per edge; lanes cover F contiguous channels.
//   F = H*C feature width (128 layer-1, 256 layer-2), H = 4 heads.
//   Lane owns CH = F/32 contiguous channels -> float4 coalesced loads.
//   Head h owns lanes [8h, 8h+8): 3-step shfl_xor reduction per head.
//   logits[e,h] = att[h,:] . leaky_relu(xl[src]+xr[dst]); segment-max(dst).
// Edges e >= E are implicit self-loops (src = dst = e - E).
// ---------------------------------------------------------------------------
template <int F>
__global__ __launch_bounds__(256) void k_edge_logits(
    const float* __restrict__ xl, const float* __restrict__ xr,
    const float* __restrict__ att,
    const int* __restrict__ src, const int* __restrict__ dst,
    float* __restrict__ elog, unsigned* __restrict__ maxbuf,
    int E, int n) {
    constexpr int CH4 = F / 128;                 // float4s per lane (1 or 2)
    const int wid  = (blockIdx.x * blockDim.x + threadIdx.x) >> 5;  // edge
    const int lane = threadIdx.x & 31;
    if (wid >= E + n) return;                    // uniform per wave
    int s, d;
    if (wid < E) { s = src[wid]; d = dst[wid]; } else { s = d = wid - E; }

    const float4* __restrict__ pl = (const float4*)(xl + (size_t)s * F) + lane * CH4;
    const float4* __restrict__ pr = (const float4*)(xr + (size_t)d * F) + lane * CH4;
    const float4* __restrict__ pa = (const float4*)att + lane * CH4;

    float partial = 0.f;
#pragma unroll
    for (int i = 0; i < CH4; ++i) {
        float4 a = pl[i], b = pr[i], w = pa[i];
        float v;
        v = a.x + b.x; v = v > 0.f ? v : 0.2f * v; partial += w.x * v;
        v = a.y + b.y; v = v > 0.f ? v : 0.2f * v; partial += w.y * v;
        v = a.z + b.z; v = v > 0.f ? v : 0.2f * v; partial += w.z * v;
        v = a.w + b.w; v = v > 0.f ? v : 0.2f * v; partial += w.w * v;
    }
    // reduce within each 8-lane head group
    partial += __shfl_xor(partial, 1, 32);
    partial += __shfl_xor(partial, 2, 32);
    partial += __shfl_xor(partial, 4, 32);

    if ((lane & 7) == 0) {
        int h = lane >> 3;
        elog[(size_t)wid * 4 + h] = partial;
        atomicMax(maxbuf + (size_t)d * 4 + h, fenc(partial));
    }
}

// ---------------------------------------------------------------------------
// Edge pass B: ex = exp(logit - m[dst]); denom[dst] += ex  (ex stored in-place)
// One thread per (edge, head).
// ---------------------------------------------------------------------------
__global__ void k_edge_exp(const int* __restrict__ dst,
                           float* __restrict__ elog,
                           const unsigned* __restrict__ maxbuf,
                           float* __restrict__ denom,
                           int E, int n) {
    int tid = blockIdx.x * blockDim.x + threadIdx.x;
    if (tid >= (E + n) * 4) return;
    int h = tid & 3;
    int e = tid >> 2;
    int d = (e < E) ? dst[e] : e - E;
    float m  = fdec(maxbuf[(size_t)d * 4 + h]);
    float ex = __expf(elog[tid] - m);
    elog[tid] = ex;
    atomicAdd(denom + (size_t)d * 4 + h, ex);
}

// ---------------------------------------------------------------------------
// Edge pass C: one WAVE per edge; lanes cover contiguous channels.
//   acc[dst,h,:] += (ex / denom[dst,h]) * xl[src,h,:]
//   Coalesced gathers and cacheline-dense atomic bursts.
// ---------------------------------------------------------------------------
template <int F>
__global__ __launch_bounds__(256) void k_edge_aggr(
    const float* __restrict__ xl,
    const float* __restrict__ elog, const float* __restrict__ denom,
    const int* __restrict__ src, const int* __restrict__ dst,
    float* __restrict__ acc, int E, int n) {
    constexpr int CH = F / 32;                   // channels per lane (4 or 8)
    const int wid  = (blockIdx.x * blockDim.x + threadIdx.x) >> 5;  // edge
    const int lane = threadIdx.x & 31;
    if (wid >= E + n) return;
    int s, d;
    if (wid < E) { s = src[wid]; d = dst[wid]; } else { s = d = wid - E; }

    const int h = lane >> 3;                     // head owning this lane
    float alpha = elog[(size_t)wid * 4 + h] / denom[(size_t)d * 4 + h];

    const float* __restrict__ pl = xl + (size_t)s * F + lane * CH;
    float* __restrict__ po = acc + (size_t)d * F + lane * CH;
#pragma unroll
    for (int c = 0; c < CH; ++c)
        atomicAdd(po + c, alpha * pl[c]);
}

// ---------------------------------------------------------------------------
// Post layer 1: y1 = relu(acc1 + bias1)   (concat: [N, 4*32] contiguous)
// ---------------------------------------------------------------------------
__global__ void k_post1(const float* __restrict__ acc,
                        const float* __restrict__ bias,
                        float* __restrict__ y, int total) {
    int i = blockIdx.x * blockDim.x + threadIdx.x;
    if (i >= total) return;
    float v = acc[i] + bias[i & 127];
    y[i] = v > 0.f ? v : 0.f;
}

// ---------------------------------------------------------------------------
// Post layer 2: y2 = relu(mean_h(acc2) + bias2)   ([N,4,64] -> [N,64])
// ---------------------------------------------------------------------------
__global__ void k_post2(const float* __restrict__ acc,
                        const float* __restrict__ bias,
                        float* __restrict__ y, int total) {
    int i = blockIdx.x * blockDim.x + threadIdx.x;
    if (i >= total) return;
    int c = i & 63;
    int node = i >> 6;
    const float* __restrict__ p = acc + (size_t)node * 256 + c;
    float v = 0.25f * (p[0] + p[64] + p[128] + p[192]) + bias[c];
    y[i] = v > 0.f ? v : 0.f;
}

// ---------------------------------------------------------------------------
// Final head: out = sigmoid(y2 @ W3 + b3)   (64 -> 5, too skinny for WMMA)
// ---------------------------------------------------------------------------
__global__ void k_final(const float* __restrict__ y2,
                        const float* __restrict__ W3,
                        const float* __restrict__ b3,
                        float* __restrict__ out, int total) {
    int i = blockIdx.x * blockDim.x + threadIdx.x;
    if (i >= total) return;
    int j = i % 5;
    int node = i / 5;
    const float* __restrict__ p = y2 + (size_t)node * 64;
    float s = b3[j];
#pragma unroll
    for (int k = 0; k < 64; ++k) s += p[k] * W3[k * 5 + j];
    out[i] = 1.f / (1.f + __expf(-s));
}

// ---------------------------------------------------------------------------
extern "C" void kernel_launch(void* const* d_in, const int* in_sizes, int n_in,
                              void* d_out, int out_size, void* d_ws, size_t ws_size,
                              hipStream_t stream) {
    const float* x     = (const float*)d_in[0];
    const int*   ei    = (const int*)  d_in[1];
    const float* lb    = (const float*)d_in[2];
    const float* ub    = (const float*)d_in[3];
    const float* W1l   = (const float*)d_in[4];
    const float* b1l   = (const float*)d_in[5];
    const float* W1r   = (const float*)d_in[6];
    const float* b1r   = (const float*)d_in[7];
    const float* att1  = (const float*)d_in[8];
    const float* bias1 = (const float*)d_in[9];
    const float* W2l   = (const float*)d_in[10];
    const float* b2l   = (const float*)d_in[11];
    const float* W2r   = (const float*)d_in[12];
    const float* b2r   = (const float*)d_in[13];
    const float* att2  = (const float*)d_in[14];
    const float* bias2 = (const float*)d_in[15];
    const float* W3    = (const float*)d_in[16];
    const float* b3    = (const float*)d_in[17];

    const int n  = in_sizes[0] / 16;   // 30000 (multiple of 16)
    const int E  = in_sizes[1] / 2;    // 480000
    const int* src = ei;               // edge_index row 0
    const int* dst = ei + E;           // edge_index row 1
    const size_t nz = (size_t)n;

    // Workspace layout (floats), with lifetime-based aliasing (~103 MB):
    //   [0,        n*16)   y0        (dead after layer-1 GEMMs)
    //   [n*16,     n*144)  xl1       (dead after layer-1 pass C)
    //   [n*144,    n*272)  xr1       (dead after layer-1 pass A)
    //   [n*272,    n*400)  acc1      (dead after post1)
    //   [n*400,    n*528)  y1        (dead after layer-2 GEMMs)
    //   [0,        n*256)  xl2       (reuses y0/xl1/xr1 region)
    //   [n*256,    n*512)  acc2      (reuses acc1 + y1 region; memset AFTER GEMMs)
    //   [n*528,    n*784)  xr2       (dead after layer-2 pass A)
    //   [n*528,    n*592)  y2        (reuses xr2 region)
    //   [n*784,    n*784 + Et*4)  elog/ex
    //   then: maxbuf (n*4 uints), denom (n*4 floats)
    float* ws = (float*)d_ws;
    float*    y0     = ws;
    float*    xl1    = ws + nz * 16;
    float*    xr1    = xl1 + nz * 128;
    float*    acc1   = xr1 + nz * 128;
    float*    y1     = acc1 + nz * 128;
    float*    xl2    = ws;
    float*    acc2   = ws + nz * 256;
    float*    xr2    = ws + nz * 528;
    float*    y2     = ws + nz * 528;
    float*    elog   = ws + nz * 784;
    unsigned* maxbuf = (unsigned*)(elog + (size_t)(E + n) * 4);
    float*    denom  = (float*)(maxbuf + nz * 4);

    const int TB = 256;
    const int Et = E + n;
    const int wgrid = (Et * 32 + TB - 1) / TB;     // wave-per-edge kernels
    const int egrid = (Et * 4 + TB - 1) / TB;      // thread-per-(edge,head)
    const dim3 gblk(32, 4);

    // ---- normalize ----
    k_norm<<<(n * 16 + TB - 1) / TB, TB, 0, stream>>>(x, lb, ub, y0, n * 16);

    // ---- layer 1: GEMMs (fp32 WMMA) ----
    k_gemm_wmma<<<dim3(128 / 64, n / 16), gblk, 0, stream>>>(y0, W1l, b1l, xl1, 16, 128);
    k_gemm_wmma<<<dim3(128 / 64, n / 16), gblk, 0, stream>>>(y0, W1r, b1r, xr1, 16, 128);

    // ---- layer 1: edge softmax + aggregate ----
    (void)hipMemsetAsync(maxbuf, 0, nz * 4 * sizeof(unsigned), stream);
    (void)hipMemsetAsync(denom,  0, nz * 4 * sizeof(float),    stream);
    (void)hipMemsetAsync(acc1,   0, nz * 128 * sizeof(float),  stream);
    k_edge_logits<128><<<wgrid, TB, 0, stream>>>(xl1, xr1, att1, src, dst, elog, maxbuf, E, n);
    k_edge_exp<<<egrid, TB, 0, stream>>>(dst, elog, maxbuf, denom, E, n);
    k_edge_aggr<128><<<wgrid, TB, 0, stream>>>(xl1, elog, denom, src, dst, acc1, E, n);
    k_post1<<<(n * 128 + TB - 1) / TB, TB, 0, stream>>>(acc1, bias1, y1, n * 128);

    // ---- layer 2: GEMMs (fp32 WMMA) ----
    k_gemm_wmma<<<dim3(256 / 64, n / 16), gblk, 0, stream>>>(y1, W2l, b2l, xl2, 128, 256);
    k_gemm_wmma<<<dim3(256 / 64, n / 16), gblk, 0, stream>>>(y1, W2r, b2r, xr2, 128, 256);

    // ---- layer 2: edge softmax + aggregate ----
    (void)hipMemsetAsync(maxbuf, 0, nz * 4 * sizeof(unsigned), stream);
    (void)hipMemsetAsync(denom,  0, nz * 4 * sizeof(float),    stream);
    (void)hipMemsetAsync(acc2,   0, nz * 256 * sizeof(float),  stream);
    k_edge_logits<256><<<wgrid, TB, 0, stream>>>(xl2, xr2, att2, src, dst, elog, maxbuf, E, n);
    k_edge_exp<<<egrid, TB, 0, stream>>>(dst, elog, maxbuf, denom, E, n);
    k_edge_aggr<256><<<wgrid, TB, 0, stream>>>(xl2, elog, denom, src, dst, acc2, E, n);
    k_post2<<<(n * 64 + TB - 1) / TB, TB, 0, stream>>>(acc2, bias2, y2, n * 64);

    // ---- final linear + sigmoid ----
    k_final<<<(n * 5 + TB - 1) / TB, TB, 0, stream>>>(y2, W3, b3, (float*)d_out, n * 5);
}